// Attention_69810398429817
// MI455X (gfx1250) — compile-verified
//
#include <hip/hip_runtime.h>
#include <hip/hip_bf16.h>
#include <math.h>
#include <stdint.h>

// Problem constants (from reference)
#define BB    8
#define TQ    128
#define TV    256
#define DD    256
#define UU    128
#define LN_EPS 1e-3f

typedef float v2f __attribute__((ext_vector_type(2)));
typedef float v8f __attribute__((ext_vector_type(8)));

// ---------------------------------------------------------------------------
// Kernel 1: projection GEMM  out[r,u] = sum_k X[r,k]*W[k,u] + bias[u]
//   One wave computes one 16x16 tile of out via V_WMMA_F32_16X16X4_F32.
//   A (16x4 f32): lanes 0-15 -> {K0,K1}, lanes 16-31 -> {K2,K3}, M=lane&15
//   B (4x16 f32): mirrored; N=lane&15
//   C/D (16x16 f32): VGPR i -> M = i + 8*(lane>=16), N = lane&15
// ---------------------------------------------------------------------------
__global__ void proj_gemm(const float* __restrict__ X, const float* __restrict__ W,
                          const float* __restrict__ bias, float* __restrict__ out,
                          int R, int K, int U) {
  int wave = (blockIdx.x * blockDim.x + threadIdx.x) >> 5;
  int lane = threadIdx.x & 31;
  int tilesU = U >> 4;
  int rt = wave / tilesU;
  int ct = wave - rt * tilesU;
  int row0 = rt << 4, col0 = ct << 4;

  int mn  = lane & 15;
  int hi  = lane >> 4;
  int kS  = hi * 2;

  const float* Arow = X + (size_t)(row0 + mn) * K;
  const float* Bcol = W + col0 + mn;

  v8f c = {0.f,0.f,0.f,0.f,0.f,0.f,0.f,0.f};
  #pragma unroll 4
  for (int k0 = 0; k0 < K; k0 += 4) {
    v2f a, b;
    a.x = Arow[k0 + kS];
    a.y = Arow[k0 + kS + 1];
    const float* bp = Bcol + (size_t)(k0 + kS) * U;
    b.x = bp[0];
    b.y = bp[U];
    c = __builtin_amdgcn_wmma_f32_16x16x4_f32(false, a, false, b,
                                              (short)0, c, false, false);
  }

  float bv = bias[col0 + mn];
  int mrow = hi * 8;
  #pragma unroll
  for (int i = 0; i < 8; ++i) {
    out[(size_t)(row0 + mrow + i) * U + col0 + mn] = c[i] + bv;
  }
}

// ---------------------------------------------------------------------------
// Kernel 2: additive scores + masked softmax.  One block per (b,q) row,
// 256 threads; thread t owns value-position v = t.
//   score[v] = sum_u tanh(qa[b,q,u] + ka[b,v,u]) * Vw[u] + Vb
// k_attn is staged into LDS *transposed* via per-lane async global->LDS
// copies (GLOBAL_LOAD_ASYNC_TO_LDS_B32, ASYNCcnt), double-buffered so the
// DMA of tile t+1 overlaps the tanh (TRANS pipe) compute of tile t.
// ---------------------------------------------------------------------------
#define TU 16                    // u-rows per tile
#define NTILE (UU / TU)          // 8 tiles
#define KSPITCH (TV + 1)         // +1 pad -> conflict-free reads

__global__ void scores_softmax(const float* __restrict__ qa,   // [B*TQ, U]
                               const float* __restrict__ ka,   // [B*TV, U]
                               const float* __restrict__ Vw,   // [U]
                               const float* __restrict__ Vb,   // [1]
                               const int*   __restrict__ vmask,// [B, TV]
                               float* __restrict__ wout) {     // [B*TQ, TV]
  __shared__ float qs[UU];
  __shared__ float vs[UU];
  __shared__ float ks[2][TU][KSPITCH];   // 2 * 16 * 257 * 4B ~= 32.9 KB
  __shared__ float red[256];

  int bq  = blockIdx.x;            // 0..B*TQ-1
  int b   = bq / TQ;
  int tid = threadIdx.x;           // == v

  if (tid < UU) {
    qs[tid] = qa[(size_t)bq * UU + tid];
    vs[tid] = Vw[tid];
  }

  const float* kbase = ka + (size_t)b * TV * UU;

  // issue one tile's async copies: ks[buf][uu][v] = ka[b, v, u0+uu]
  // per-lane src AND dst -> free transpose.  16 async ops per wave per tile.
  auto issue_tile = [&](int u0, int buf) {
    #pragma unroll
    for (int j = 0; j < (TU * TV) / 256; ++j) {   // 16 iters
      int i  = j * 256 + tid;
      int v  = i >> 4;                            // i / TU-group: v = i/16? no:
      // mapping: consecutive lanes -> consecutive uu within one v-row
      v      = i >> 4;          // 16 uu per v  -> 16 contiguous floats per row
      int uu = i & 15;
      const float*  src = kbase + (size_t)v * UU + u0 + uu;
      uint32_t dst = (uint32_t)(uintptr_t)&ks[buf][uu][v];
      asm volatile("global_load_async_to_lds_b32 %0, %1, off"
                   :: "v"(dst), "v"(src)
                   : "memory");
    }
  };

  issue_tile(0, 0);

  float s = 0.f;
  for (int t = 0; t < NTILE; ++t) {
    int u0  = t * TU;
    int buf = t & 1;
    if (t + 1 < NTILE) {
      issue_tile(u0 + TU, buf ^ 1);
      // our tile-t copies (16 ops) complete before the 16 just issued
      asm volatile("s_wait_asynccnt 16" ::: "memory");
    } else {
      asm volatile("s_wait_asynccnt 0" ::: "memory");
    }
    __syncthreads();             // all waves' tile-t copies visible (+qs/vs)
    #pragma unroll
    for (int uu = 0; uu < TU; ++uu) {
      s += tanhf(qs[u0 + uu] + ks[buf][uu][tid]) * vs[u0 + uu];
    }
    __syncthreads();             // compute done before buffer is re-filled
  }
  s += Vb[0];
  if (vmask[b * TV + tid] == 0) s -= 1e9f;

  // max-reduce
  red[tid] = s; __syncthreads();
  for (int off = 128; off > 0; off >>= 1) {
    if (tid < off) red[tid] = fmaxf(red[tid], red[tid + off]);
    __syncthreads();
  }
  float mx = red[0]; __syncthreads();

  float e = __expf(s - mx);
  red[tid] = e; __syncthreads();
  for (int off = 128; off > 0; off >>= 1) {
    if (tid < off) red[tid] += red[tid + off];
    __syncthreads();
  }
  float inv = 1.f / red[0];

  wout[(size_t)bq * TV + tid] = e * inv;
}

// ---------------------------------------------------------------------------
// Kernel 3: attn_out = weights @ value  (per batch 128x256 @ 256x256),
// fused residual add of query.  WMMA f32 16x16x4, one 16x16 tile per wave.
// ---------------------------------------------------------------------------
__global__ void attn_gemm(const float* __restrict__ Wt,  // [B, TQ, TV]
                          const float* __restrict__ V,   // [B, TV, DD]
                          const float* __restrict__ Q,   // [B, TQ, DD]
                          float* __restrict__ Y) {       // [B, TQ, DD]
  int wave = (blockIdx.x * blockDim.x + threadIdx.x) >> 5;
  int lane = threadIdx.x & 31;
  const int tilesN = DD >> 4;                 // 16
  const int tilesM = TQ >> 4;                 // 8
  const int tilesPerBatch = tilesM * tilesN;  // 128
  int b  = wave / tilesPerBatch;
  int t  = wave - b * tilesPerBatch;
  int rt = t / tilesN;
  int ct = t - rt * tilesN;
  int row0 = rt << 4, col0 = ct << 4;

  int mn = lane & 15;
  int hi = lane >> 4;
  int kS = hi * 2;

  const float* Arow = Wt + ((size_t)b * TQ + row0 + mn) * TV;
  const float* Bcol = V + (size_t)b * TV * DD + col0 + mn;

  v8f c = {0.f,0.f,0.f,0.f,0.f,0.f,0.f,0.f};
  #pragma unroll 4
  for (int k0 = 0; k0 < TV; k0 += 4) {
    v2f a, bb;
    a.x = Arow[k0 + kS];
    a.y = Arow[k0 + kS + 1];
    const float* bp = Bcol + (size_t)(k0 + kS) * DD;
    bb.x = bp[0];
    bb.y = bp[DD];
    c = __builtin_amdgcn_wmma_f32_16x16x4_f32(false, a, false, bb,
                                              (short)0, c, false, false);
  }

  int mrow = hi * 8;
  #pragma unroll
  for (int i = 0; i < 8; ++i) {
    size_t idx = ((size_t)b * TQ + row0 + mrow + i) * DD + col0 + mn;
    Y[idx] = c[i] + Q[idx];
  }
}

// ---------------------------------------------------------------------------
// Kernel 4: LayerNorm over last dim (256).  One block (256 thr) per row.
// ---------------------------------------------------------------------------
__global__ void layernorm_k(const float* __restrict__ Y,
                            const float* __restrict__ gamma,
                            const float* __restrict__ beta,
                            float* __restrict__ out) {
  __shared__ float red[256];
  int row = blockIdx.x;
  int tid = threadIdx.x;

  float x = Y[(size_t)row * DD + tid];
  red[tid] = x; __syncthreads();
  for (int off = 128; off > 0; off >>= 1) {
    if (tid < off) red[tid] += red[tid + off];
    __syncthreads();
  }
  float mu = red[0] * (1.0f / DD); __syncthreads();

  float dx = x - mu;
  red[tid] = dx * dx; __syncthreads();
  for (int off = 128; off > 0; off >>= 1) {
    if (tid < off) red[tid] += red[tid + off];
    __syncthreads();
  }
  float var = red[0] * (1.0f / DD);
  float r = rsqrtf(var + LN_EPS);
  out[(size_t)row * DD + tid] = dx * r * gamma[tid] + beta[tid];
}

// ---------------------------------------------------------------------------
extern "C" void kernel_launch(void* const* d_in, const int* in_sizes, int n_in,
                              void* d_out, int out_size, void* d_ws, size_t ws_size,
                              hipStream_t stream) {
  (void)in_sizes; (void)n_in; (void)out_size; (void)ws_size;

  const float* query = (const float*)d_in[0];   // [B,TQ,D]
  const float* value = (const float*)d_in[1];   // [B,TV,D]
  const int*   vmask = (const int*)d_in[2];     // [B,TV] bool->int
  const float* Wq_w  = (const float*)d_in[3];   // [D,U]
  const float* Wq_b  = (const float*)d_in[4];   // [U]
  const float* Wk_w  = (const float*)d_in[5];   // [D,U]
  const float* Wk_b  = (const float*)d_in[6];   // [U]
  const float* V_w   = (const float*)d_in[7];   // [U,1]
  const float* V_b   = (const float*)d_in[8];   // [1]
  const float* g     = (const float*)d_in[9];   // [D]
  const float* be    = (const float*)d_in[10];  // [D]

  float* result  = (float*)d_out;                        // [B,TQ,D]
  float* weights = (float*)d_out + (size_t)BB * TQ * DD; // [B,TQ,TV]

  float* qa = (float*)d_ws;                  // [B*TQ, U]
  float* ka = qa + (size_t)BB * TQ * UU;     // [B*TV, U]
  float* y  = ka + (size_t)BB * TV * UU;     // [B*TQ, D]

  proj_gemm<<<64, 256, 0, stream>>>(query, Wq_w, Wq_b, qa, BB * TQ, DD, UU);
  proj_gemm<<<128, 256, 0, stream>>>(value, Wk_w, Wk_b, ka, BB * TV, DD, UU);
  scores_softmax<<<BB * TQ, 256, 0, stream>>>(qa, ka, V_w, V_b, vmask, weights);
  attn_gemm<<<128, 256, 0, stream>>>(weights, value, query, y);
  layernorm_k<<<BB * TQ, 256, 0, stream>>>(y, g, be, result);
}